// model_48644799595116
// MI455X (gfx1250) — compile-verified
//
#include <hip/hip_runtime.h>
#include <hip/hip_bf16.h>

typedef float v2f __attribute__((ext_vector_type(2)));
typedef float v8f __attribute__((ext_vector_type(8)));

#define NROW 4000
#define TSTEPS 16

// ---------------------------------------------------------------------------
// Generic GEMM: Y[r,c] = act( bias[c] + sum_i X[r,i] * W[i,c] )
// act: 0 = none, 1 = relu, 2 = sigmoid
// ---------------------------------------------------------------------------
__global__ void gemm_act(const float* __restrict__ X, const float* __restrict__ W,
                         const float* __restrict__ bias, float* __restrict__ Y,
                         int rows, int K, int C, int act) {
  int idx = blockIdx.x * blockDim.x + threadIdx.x;
  if (idx >= rows * C) return;
  int r = idx / C, c = idx % C;
  float s = bias ? bias[c] : 0.0f;
  const float* x = X + (size_t)r * K;
  for (int i = 0; i < K; ++i) s = fmaf(x[i], W[i * C + c], s);
  if (act == 1) s = fmaxf(s, 0.0f);
  else if (act == 2) s = 1.0f / (1.0f + __expf(-s));
  Y[idx] = s;
}

// ---------------------------------------------------------------------------
// Per-column batchnorm statistics over rows; fold gamma/beta into scale/shift:
//   bn(x) = x*scale[c] + shift[c]
// ---------------------------------------------------------------------------
__global__ void colstats(const float* __restrict__ X, int rows, int C,
                         const float* __restrict__ g, const float* __restrict__ b,
                         float* __restrict__ sc, float* __restrict__ sh) {
  __shared__ float s1[256], s2[256];
  int c = blockIdx.x;
  float a = 0.0f, q = 0.0f;
  for (int r = threadIdx.x; r < rows; r += blockDim.x) {
    float v = X[(size_t)r * C + c];
    a += v; q += v * v;
  }
  s1[threadIdx.x] = a; s2[threadIdx.x] = q;
  __syncthreads();
  for (int o = blockDim.x >> 1; o > 0; o >>= 1) {
    if (threadIdx.x < o) { s1[threadIdx.x] += s1[threadIdx.x + o]; s2[threadIdx.x] += s2[threadIdx.x + o]; }
    __syncthreads();
  }
  if (threadIdx.x == 0) {
    float m   = s1[0] / (float)rows;
    float var = fmaxf(s2[0] / (float)rows - m * m, 0.0f);
    float rs  = rsqrtf(var + 1e-5f);
    float scale = g[c] * rs;
    sc[c] = scale;
    sh[c] = b[c] - m * scale;
  }
}

// ---------------------------------------------------------------------------
// Fused  Y = act( relu(bn(X)) @ W + bias ).  BN+ReLU recomputed inline.
// If trans: Y stored column-major [Cpad][rows] with zero padding for c >= C.
// ---------------------------------------------------------------------------
__global__ void bn_gemm_act(const float* __restrict__ X, const float* __restrict__ sc,
                            const float* __restrict__ sh,
                            const float* __restrict__ W, const float* __restrict__ bias,
                            float* __restrict__ Y,
                            int rows, int K, int C, int Cpad, int act, int trans) {
  int idx = blockIdx.x * blockDim.x + threadIdx.x;
  if (idx >= rows * Cpad) return;
  int r = idx / Cpad, c = idx % Cpad;
  float s = 0.0f;
  if (c < C) {
    s = bias ? bias[c] : 0.0f;
    const float* x = X + (size_t)r * K;
    for (int i = 0; i < K; ++i) {
      float v = fmaxf(fmaf(x[i], sc[i], sh[i]), 0.0f);
      s = fmaf(v, W[i * C + c], s);
    }
    if (act == 1) s = fmaxf(s, 0.0f);
    else if (act == 2) s = 1.0f / (1.0f + __expf(-s));
  }
  if (trans) Y[(size_t)c * rows + r] = s;
  else       Y[(size_t)r * C + c]    = s;
}

// ---------------------------------------------------------------------------
// h = Ar @ z  via V_WMMA_F32_16X16X4_F32.
// Block = 8 waves; one 16-row M-tile per block; wave w handles K in
// [w*500, w*500+500); three 16x16 accumulators cover 48 (=40 padded) columns.
// A frag (16x4 f32): lane L<16 -> Ar[m0+L][k..k+1], lanes 16..31 -> [k+2..k+3].
// B frag (4x16 f32): zT stored [48][4000] so the same contiguous v2f pattern
// applies: lane n<16 -> z[k..k+1][n], lanes 16..31 -> z[k+2..k+3][n].
// ---------------------------------------------------------------------------
__global__ __launch_bounds__(256) void wmma_h_kernel(const float* __restrict__ Ar,
                                                     const float* __restrict__ zT,
                                                     float* __restrict__ h) {
  __shared__ float red[8][32][24];
  const int m0   = blockIdx.x * 16;
  const int lane = threadIdx.x & 31;
  const int wv   = threadIdx.x >> 5;
  const int hl   = lane >> 4;       // half: 0 or 1
  const int l15  = lane & 15;
  const int kOff = hl * 2;

  const float* arow = Ar + (size_t)(m0 + l15) * NROW + kOff;
  const float* b0p  = zT + (size_t)( 0 + l15) * NROW + kOff;
  const float* b1p  = zT + (size_t)(16 + l15) * NROW + kOff;
  const float* b2p  = zT + (size_t)(32 + l15) * NROW + kOff;

  v8f acc0 = {}, acc1 = {}, acc2 = {};
  const int k0 = wv * 500;
  for (int k = k0; k < k0 + 500; k += 4) {
    v2f a  = *(const v2f*)(arow + k);
    v2f b0 = *(const v2f*)(b0p + k);
    v2f b1 = *(const v2f*)(b1p + k);
    v2f b2 = *(const v2f*)(b2p + k);
    acc0 = __builtin_amdgcn_wmma_f32_16x16x4_f32(false, a, false, b0, (short)0, acc0, false, false);
    acc1 = __builtin_amdgcn_wmma_f32_16x16x4_f32(false, a, false, b1, (short)0, acc1, false, false);
    acc2 = __builtin_amdgcn_wmma_f32_16x16x4_f32(false, a, false, b2, (short)0, acc2, false, false);
  }

  float* dst = &red[wv][lane][0];
#pragma unroll
  for (int i = 0; i < 8; ++i) { dst[i] = acc0[i]; dst[8 + i] = acc1[i]; dst[16 + i] = acc2[i]; }
  __syncthreads();

  if (wv == 0) {
#pragma unroll
    for (int w = 1; w < 8; ++w) {
      const float* s = &red[w][lane][0];
#pragma unroll
      for (int i = 0; i < 8; ++i) { acc0[i] += s[i]; acc1[i] += s[8 + i]; acc2[i] += s[16 + i]; }
    }
    // D layout: VGPR r, lanes 0-15 -> M=r, lanes 16-31 -> M=r+8; N = lane&15.
    const int rowBase = m0 + hl * 8;
#pragma unroll
    for (int r = 0; r < 8; ++r) {
      const int row = rowBase + r;
      h[(size_t)row * 40 + ( 0 + l15)] = acc0[r];
      h[(size_t)row * 40 + (16 + l15)] = acc1[r];
      if (32 + l15 < 40) h[(size_t)row * 40 + (32 + l15)] = acc2[r];
    }
  }
}

// ---------------------------------------------------------------------------
// hm = concat(h, md)  -> (N, 41)
// ---------------------------------------------------------------------------
__global__ void concat_hm(const float* __restrict__ h, const float* __restrict__ md,
                          float* __restrict__ hm) {
  int idx = blockIdx.x * blockDim.x + threadIdx.x;
  if (idx >= NROW * 41) return;
  int r = idx / 41, c = idx % 41;
  hm[idx] = (c < 40) ? h[(size_t)r * 40 + c] : md[r];
}

// ---------------------------------------------------------------------------
// Per-node attention mixing: Aw = A * ap5x5; x2 = Aw @ h5; out = relu(x2@Ws+bs);
// infection = relu(out[1,:] @ Wi + bi); of = out flattened.
// ---------------------------------------------------------------------------
__global__ void mix_kernel(const float* __restrict__ ap, const float* __restrict__ h,
                           const float* __restrict__ A,  const float* __restrict__ Ws,
                           const float* __restrict__ bs, const float* __restrict__ Wi,
                           const float* __restrict__ bi,
                           float* __restrict__ of, float* __restrict__ infv) {
  int n = blockIdx.x * blockDim.x + threadIdx.x;
  if (n >= NROW) return;
  float hr[40];
#pragma unroll
  for (int i = 0; i < 40; ++i) hr[i] = h[(size_t)n * 40 + i];
  float outr[5][8];
#pragma unroll
  for (int i = 0; i < 5; ++i) {
    float x2[8];
#pragma unroll
    for (int c = 0; c < 8; ++c) x2[c] = 0.0f;
#pragma unroll
    for (int j = 0; j < 5; ++j) {
      float aw = A[i * 5 + j] * ap[(size_t)n * 25 + i * 5 + j];
#pragma unroll
      for (int c = 0; c < 8; ++c) x2[c] = fmaf(aw, hr[j * 8 + c], x2[c]);
    }
#pragma unroll
    for (int c = 0; c < 8; ++c) {
      float s = bs[c];
#pragma unroll
      for (int k = 0; k < 8; ++k) s = fmaf(x2[k], Ws[k * 8 + c], s);
      outr[i][c] = fmaxf(s, 0.0f);
      of[(size_t)n * 40 + i * 8 + c] = outr[i][c];
    }
  }
  float inf = bi[0];
#pragma unroll
  for (int c = 0; c < 8; ++c) inf = fmaf(outr[1][c], Wi[c], inf);
  infv[n] = fmaxf(inf, 0.0f);
}

// ---------------------------------------------------------------------------
// d = relu(bn(t3)) @ Wd1 + bd1; emit density/md/infection outputs; build sfc.
// ---------------------------------------------------------------------------
__global__ void dens_kernel(const float* __restrict__ t3, const float* __restrict__ sc,
                            const float* __restrict__ sh, const float* __restrict__ Wd1,
                            const float* __restrict__ bd1,
                            const float* __restrict__ of, const float* __restrict__ infv,
                            float* __restrict__ dens_o, float* __restrict__ mds_o,
                            float* __restrict__ infs_o, float* __restrict__ mdbuf,
                            float* __restrict__ sfc, int t) {
  int n = blockIdx.x * blockDim.x + threadIdx.x;
  if (n >= NROW) return;
  float d0 = bd1[0], d1 = bd1[1];
  for (int i = 0; i < 32; ++i) {
    float v = fmaxf(fmaf(t3[(size_t)n * 32 + i], sc[i], sh[i]), 0.0f);
    d0 = fmaf(v, Wd1[i * 2 + 0], d0);
    d1 = fmaf(v, Wd1[i * 2 + 1], d1);
  }
  dens_o[(size_t)t * NROW + n] = d0;
  mds_o [(size_t)t * NROW + n] = d1;
  infs_o[(size_t)t * NROW + n] = infv[n];
  mdbuf[n] = d1;
  for (int i = 0; i < 40; ++i) sfc[(size_t)n * 42 + i] = of[(size_t)n * 40 + i];
  sfc[(size_t)n * 42 + 40] = d0;
  sfc[(size_t)n * 42 + 41] = infv[n];
}

// ---------------------------------------------------------------------------
extern "C" void kernel_launch(void* const* d_in, const int* in_sizes, int n_in,
                              void* d_out, int out_size, void* d_ws, size_t ws_size,
                              hipStream_t stream) {
  (void)in_sizes; (void)n_in; (void)out_size; (void)ws_size;
  const float* A    = (const float*)d_in[0];
  const float* Areg = (const float*)d_in[1];
  const float* x    = (const float*)d_in[2];
  const float* md0  = (const float*)d_in[3];
  const float* W0   = (const float*)d_in[4];
  const float* g0   = (const float*)d_in[5];
  const float* be0  = (const float*)d_in[6];
  const float* W1   = (const float*)d_in[7];
  const float* b1   = (const float*)d_in[8];
  const float* Wr0  = (const float*)d_in[9];
  const float* gr   = (const float*)d_in[10];
  const float* ber  = (const float*)d_in[11];
  const float* Wr1  = (const float*)d_in[12];
  const float* br1  = (const float*)d_in[13];
  const float* Wa0  = (const float*)d_in[14];
  const float* ba0  = (const float*)d_in[15];
  const float* ga   = (const float*)d_in[16];
  const float* bea  = (const float*)d_in[17];
  const float* Wa1  = (const float*)d_in[18];
  const float* ba1  = (const float*)d_in[19];
  const float* Ws   = (const float*)d_in[20];
  const float* bs   = (const float*)d_in[21];
  const float* Wi   = (const float*)d_in[22];
  const float* bi   = (const float*)d_in[23];
  const float* Wd0  = (const float*)d_in[24];
  const float* bd0  = (const float*)d_in[25];
  const float* gd   = (const float*)d_in[26];
  const float* bed  = (const float*)d_in[27];
  const float* Wd1  = (const float*)d_in[28];
  const float* bd1  = (const float*)d_in[29];
  const float* Wc0  = (const float*)d_in[30];
  const float* gc   = (const float*)d_in[31];
  const float* bec  = (const float*)d_in[32];
  const float* Wc1  = (const float*)d_in[33];
  const float* bc1  = (const float*)d_in[34];

  float* out   = (float*)d_out;
  float* dens  = out;                 // (16,4000)
  float* mds   = out + 64000;        // (16,4000)
  float* infs  = out + 128000;       // (16,4000)
  float* sffin = out + 192000;       // (4000,40)

  float* ws = (float*)d_ws;
  float* t1    = ws;              // up to 4000x128
  float* sf    = ws + 512000;     // 4000x40
  float* zT    = ws + 672000;     // 48x4000 (padded, transposed)
  float* hbuf  = ws + 864000;     // 4000x40
  float* hm    = ws + 1024000;    // 4000x41
  float* apb   = ws + 1188000;    // 4000x25
  float* ofb   = ws + 1288000;    // 4000x40
  float* infb  = ws + 1448000;    // 4000
  float* sfcb  = ws + 1452000;    // 4000x42
  float* mdbuf = ws + 1620000;    // 4000
  float* scb   = ws + 1624000;    // 128
  float* shb   = ws + 1624128;    // 128

  const int TB = 256;
#define GRID(n) (((n) + TB - 1) / TB)

  // ---- init: sf = sigmoid(relu(bn(x@W0)) @ W1 + b1) ----
  gemm_act<<<GRID(NROW * 128), TB, 0, stream>>>(x, W0, nullptr, t1, NROW, 8, 128, 0);
  colstats<<<128, 256, 0, stream>>>(t1, NROW, 128, g0, be0, scb, shb);
  bn_gemm_act<<<GRID(NROW * 40), TB, 0, stream>>>(t1, scb, shb, W1, b1, sf,
                                                  NROW, 128, 40, 40, 2, 0);
  hipMemcpyAsync(mdbuf, md0, NROW * sizeof(float), hipMemcpyDeviceToDevice, stream);

  for (int t = 0; t < TSTEPS; ++t) {
    const float* Ar = Areg + (size_t)t * NROW * NROW;

    // z = relu(bn(sf@Wr0)); z2 = relu(z@Wr1+br1), stored transposed+padded
    gemm_act<<<GRID(NROW * 64), TB, 0, stream>>>(sf, Wr0, nullptr, t1, NROW, 40, 64, 0);
    colstats<<<64, 256, 0, stream>>>(t1, NROW, 64, gr, ber, scb, shb);
    bn_gemm_act<<<GRID(NROW * 48), TB, 0, stream>>>(t1, scb, shb, Wr1, br1, zT,
                                                    NROW, 64, 40, 48, 1, 1);

    // h = Ar @ z  (the heavy streaming matmul, WMMA f32)
    wmma_h_kernel<<<NROW / 16, 256, 0, stream>>>(Ar, zT, hbuf);

    // attention probabilities
    concat_hm<<<GRID(NROW * 41), TB, 0, stream>>>(hbuf, mdbuf, hm);
    gemm_act<<<GRID(NROW * 32), TB, 0, stream>>>(hm, Wa0, ba0, t1, NROW, 41, 32, 0);
    colstats<<<32, 256, 0, stream>>>(t1, NROW, 32, ga, bea, scb, shb);
    bn_gemm_act<<<GRID(NROW * 25), TB, 0, stream>>>(t1, scb, shb, Wa1, ba1, apb,
                                                    NROW, 32, 25, 25, 2, 0);

    // per-node mixing + infection
    mix_kernel<<<GRID(NROW), TB, 0, stream>>>(apb, hbuf, A, Ws, bs, Wi, bi, ofb, infb);

    // density head
    gemm_act<<<GRID(NROW * 32), TB, 0, stream>>>(ofb, Wd0, bd0, t1, NROW, 40, 32, 0);
    colstats<<<32, 256, 0, stream>>>(t1, NROW, 32, gd, bed, scb, shb);
    dens_kernel<<<GRID(NROW), TB, 0, stream>>>(t1, scb, shb, Wd1, bd1, ofb, infb,
                                               dens, mds, infs, mdbuf, sfcb, t);

    // carry update: sf = relu(relu(bn(sfc@Wc0)) @ Wc1 + bc1)
    gemm_act<<<GRID(NROW * 80), TB, 0, stream>>>(sfcb, Wc0, nullptr, t1, NROW, 42, 80, 0);
    colstats<<<80, 256, 0, stream>>>(t1, NROW, 80, gc, bec, scb, shb);
    bn_gemm_act<<<GRID(NROW * 40), TB, 0, stream>>>(t1, scb, shb, Wc1, bc1, sf,
                                                    NROW, 80, 40, 40, 1, 0);
  }

  hipMemcpyAsync(sffin, sf, NROW * 40 * sizeof(float), hipMemcpyDeviceToDevice, stream);
#undef GRID
}